// GATModel_23184233464565
// MI455X (gfx1250) — compile-verified
//
#include <hip/hip_runtime.h>
#include <math.h>

// ---------------------------------------------------------------------------
// 3-layer GAT + MLP head for MI455X (gfx1250, wave32).
// GEMMs use V_WMMA_F32_16X16X4_F32 (exact fp32 matrix path), templated on
// <K, M> so the K-loop fully unrolls with constant instruction offsets.
// Edge phase: wave-per-edge gather of h[src] + global_atomic_add_f32 scatter.
// Segment softmax: float atomic-max (sign-split int/uint trick) + atomic sums.
// ---------------------------------------------------------------------------

#define HEADS 4
#define CH    64
#define HC    256
#define NEG_SLOPE 0.2f

typedef float v2f __attribute__((ext_vector_type(2)));
typedef float v8f __attribute__((ext_vector_type(8)));

__device__ __forceinline__ void atomicMaxF(float* addr, float val) {
  // works for any mix of signs; addr must be initialized to -inf
  if (val >= 0.0f) atomicMax((int*)addr, __float_as_int(val));
  else             atomicMin((unsigned int*)addr, __float_as_uint(val));
}

__device__ __forceinline__ float leaky(float v) {
  return v > 0.0f ? v : NEG_SLOPE * v;
}

// ---------------------------------------------------------------------------
// WMMA fp32 GEMM:  C[N,M] = A[N,K] @ B[K,M]  (+bias, optional ReLU)
// block = 256 threads (8 waves). Wave w computes the 16x16 tile at
// rows [16*bx, 16*bx+16), cols [128*by + 16*w, +16). A tile staged in LDS.
// K, M compile-time: full unroll, all offsets folded into load immediates.
// ---------------------------------------------------------------------------
template <int K, int M>
__global__ __launch_bounds__(256)
void wmma_gemm_f32(const float* __restrict__ A, const float* __restrict__ B,
                   float* __restrict__ Cmat, int Nrows,
                   const float* __restrict__ bias, int do_relu)
{
  constexpr int LDA = K + 4;                // pad: conflict-free column reads
  __shared__ float As[16 * LDA];
  const int r0   = blockIdx.x * 16;
  const int wave = threadIdx.x >> 5;
  const int lane = threadIdx.x & 31;
  const int c0   = blockIdx.y * 128 + wave * 16;

  // cooperative, coalesced load of the 16xK A tile into LDS (float4 chunks)
#pragma unroll
  for (int i = threadIdx.x * 4; i < 16 * K; i += 256 * 4) {
    const int row = i / K, col = i - row * K;   // col % 4 == 0
    float4 v = make_float4(0.f, 0.f, 0.f, 0.f);
    if (r0 + row < Nrows)
      v = *(const float4*)(A + (size_t)(r0 + row) * K + col);
    float* dst = &As[row * LDA + col];
    dst[0] = v.x; dst[1] = v.y; dst[2] = v.z; dst[3] = v.w;
  }
  __syncthreads();

  const int l    = lane & 15;
  const int half = lane >> 4;   // 0: K={k,k+1} / M=r ; 1: K={k+2,k+3} / M=r+8
  v8f acc = {};
  // base pointers; all further offsets are compile-time constants
  const float* __restrict__ ap = &As[l * LDA + 2 * half];   // 8B aligned
  const float* __restrict__ bp = B + (size_t)(2 * half) * M + c0 + l;

#pragma unroll
  for (int k = 0; k < K; k += 4) {
    v2f a = *(const v2f*)(ap + k);            // A 16x4: ds_load_b64, const offset
    v2f b;                                    // B 4x16: rows {ka, ka+1} over lanes
    b.x = bp[(size_t)k * M];
    b.y = bp[(size_t)(k + 1) * M];
    acc = __builtin_amdgcn_wmma_f32_16x16x4_f32(
        false, a, false, b, (short)0, acc, false, false);
  }

  const int orow = r0 + half * 8;
  const int ocol = c0 + l;
  const float bv = bias ? bias[ocol] : 0.0f;
#pragma unroll
  for (int r = 0; r < 8; ++r) {
    if (orow + r < Nrows) {
      float v = acc[r] + bv;
      if (do_relu) v = fmaxf(v, 0.0f);
      Cmat[(size_t)(orow + r) * M + ocol] = v;
    }
  }
}

// ---------------------------------------------------------------------------
// a_src[n,h] = sum_c h[n,h,c]*att_src[h,c]  (same for a_dst). Wave per (n,h).
// ---------------------------------------------------------------------------
__global__ __launch_bounds__(256)
void attn_coeff(const float* __restrict__ h, const float* __restrict__ att_src,
                const float* __restrict__ att_dst, float* __restrict__ a_src,
                float* __restrict__ a_dst, int N)
{
  const int gw   = (blockIdx.x * blockDim.x + threadIdx.x) >> 5;
  const int lane = threadIdx.x & 31;
  if (gw >= N * HEADS) return;
  const int n = gw >> 2, hd = gw & 3;

  float2 hv = *(const float2*)(h + (size_t)n * HC + hd * CH + lane * 2);
  float2 as = *(const float2*)(att_src + hd * CH + lane * 2);
  float2 ad = *(const float2*)(att_dst + hd * CH + lane * 2);
  float ps = hv.x * as.x + hv.y * as.y;
  float pd = hv.x * ad.x + hv.y * ad.y;
#pragma unroll
  for (int off = 16; off > 0; off >>= 1) {
    ps += __shfl_down(ps, off, 32);
    pd += __shfl_down(pd, off, 32);
  }
  if (lane == 0) { a_src[gw] = ps; a_dst[gw] = pd; }
}

// ---------------------------------------------------------------------------
// init kernels
// ---------------------------------------------------------------------------
__global__ void init_attn_buffers(float* __restrict__ m, float* __restrict__ s, int n)
{
  int t = blockIdx.x * blockDim.x + threadIdx.x;
  if (t < n) { m[t] = -__builtin_huge_valf(); s[t] = 0.0f; }
}

__global__ void init_out_bias(float* __restrict__ out, const float* __restrict__ bias, int total)
{
  int t = blockIdx.x * blockDim.x + threadIdx.x;
  if (t < total) out[t] = bias ? bias[t & (HC - 1)] : 0.0f;
}

// ---------------------------------------------------------------------------
// edge passes: thread per (edge, head). Self-loops are edges e >= E.
// ---------------------------------------------------------------------------
__device__ __forceinline__ void edge_sd(const int* __restrict__ ei, int e, int E,
                                        int& s, int& d)
{
  if (e < E) { s = ei[e]; d = ei[E + e]; } else { s = d = e - E; }
}

__global__ __launch_bounds__(256)
void edge_max(const int* __restrict__ ei, int E, int Etot,
              const float* __restrict__ a_src, const float* __restrict__ a_dst,
              float* __restrict__ m)
{
  int t = blockIdx.x * blockDim.x + threadIdx.x;
  if (t >= Etot * HEADS) return;
  int e = t >> 2, hd = t & 3, s, d;
  edge_sd(ei, e, E, s, d);
  float ev = leaky(a_src[s * HEADS + hd] + a_dst[d * HEADS + hd]);
  atomicMaxF(&m[d * HEADS + hd], ev);
}

__global__ __launch_bounds__(256)
void edge_expsum(const int* __restrict__ ei, int E, int Etot,
                 const float* __restrict__ a_src, const float* __restrict__ a_dst,
                 const float* __restrict__ m, float* __restrict__ ssum)
{
  int t = blockIdx.x * blockDim.x + threadIdx.x;
  if (t >= Etot * HEADS) return;
  int e = t >> 2, hd = t & 3, s, d;
  edge_sd(ei, e, E, s, d);
  float ev = leaky(a_src[s * HEADS + hd] + a_dst[d * HEADS + hd]);
  float mv = m[d * HEADS + hd];
  if (!isfinite(mv)) mv = 0.0f;
  atomicAdd(&ssum[d * HEADS + hd], __expf(ev - mv));
}

// wave per edge; lane l handles 8 channels [8l, 8l+8) of the 256-wide feature
__global__ __launch_bounds__(256)
void edge_aggregate(const int* __restrict__ ei, int E, int Etot,
                    const float* __restrict__ a_src, const float* __restrict__ a_dst,
                    const float* __restrict__ m, const float* __restrict__ ssum,
                    const float* __restrict__ h, float* __restrict__ out)
{
  const int gw   = (blockIdx.x * blockDim.x + threadIdx.x) >> 5;
  const int lane = threadIdx.x & 31;
  if (gw >= Etot) return;
  int s, d;
  edge_sd(ei, gw, E, s, d);

  const int hd = lane >> 3;                 // 8 lanes per head
  float ev = leaky(a_src[s * HEADS + hd] + a_dst[d * HEADS + hd]);
  float mv = m[d * HEADS + hd];
  if (!isfinite(mv)) mv = 0.0f;
  float alpha = __expf(ev - mv) / (ssum[d * HEADS + hd] + 1e-16f);

  const float4* hp = (const float4*)(h + (size_t)s * HC + lane * 8);
  float4 v0 = hp[0], v1 = hp[1];
  float* op = out + (size_t)d * HC + lane * 8;
  atomicAdd(op + 0, v0.x * alpha); atomicAdd(op + 1, v0.y * alpha);
  atomicAdd(op + 2, v0.z * alpha); atomicAdd(op + 3, v0.w * alpha);
  atomicAdd(op + 4, v1.x * alpha); atomicAdd(op + 5, v1.y * alpha);
  atomicAdd(op + 6, v1.z * alpha); atomicAdd(op + 7, v1.w * alpha);
}

// ---------------------------------------------------------------------------
// layer-2 head mean + bias:  y[n,c] = mean_h agg[n,h,c] + b2[c]
// ---------------------------------------------------------------------------
__global__ void head_mean_bias(const float* __restrict__ agg, const float* __restrict__ b2,
                               float* __restrict__ y, int N)
{
  int t = blockIdx.x * blockDim.x + threadIdx.x;
  if (t >= N * CH) return;
  int n = t / CH, c = t - n * CH;
  const float* p = agg + (size_t)n * HC + c;
  y[t] = 0.25f * (p[0] + p[CH] + p[2 * CH] + p[3 * CH]) + b2[c];
}

// final: out[n,j] = z[n,:] @ lw1[:,j] + lb1[j]   (z: [N,128], lw1: [128,10])
__global__ void final_linear(const float* __restrict__ z, const float* __restrict__ lw1,
                             const float* __restrict__ lb1, float* __restrict__ out, int N)
{
  int t = blockIdx.x * blockDim.x + threadIdx.x;
  if (t >= N * 10) return;
  int n = t / 10, j = t - n * 10;
  const float* zr = z + (size_t)n * 128;
  float acc = lb1[j];
#pragma unroll
  for (int k = 0; k < 128; ++k) acc += zr[k] * lw1[k * 10 + j];
  out[t] = acc;
}

// ---------------------------------------------------------------------------
extern "C" void kernel_launch(void* const* d_in, const int* in_sizes, int n_in,
                              void* d_out, int out_size, void* d_ws, size_t ws_size,
                              hipStream_t stream)
{
  const float* x   = (const float*)d_in[0];
  const int*   ei  = (const int*)d_in[1];
  const float* W0  = (const float*)d_in[2];
  const float* as0 = (const float*)d_in[3];
  const float* ad0 = (const float*)d_in[4];
  const float* b0  = (const float*)d_in[5];
  const float* W1  = (const float*)d_in[6];
  const float* as1 = (const float*)d_in[7];
  const float* ad1 = (const float*)d_in[8];
  const float* b1  = (const float*)d_in[9];
  const float* W2  = (const float*)d_in[10];
  const float* as2 = (const float*)d_in[11];
  const float* ad2 = (const float*)d_in[12];
  const float* b2  = (const float*)d_in[13];
  const float* lw0 = (const float*)d_in[14];
  const float* lb0 = (const float*)d_in[15];
  const float* lw1 = (const float*)d_in[16];
  const float* lb1 = (const float*)d_in[17];

  const int N    = in_sizes[0] / 128;
  const int E    = in_sizes[1] / 2;
  const int Etot = E + N;

  // workspace layout (floats)
  float* ws    = (float*)d_ws;
  float* h_lin = ws;  ws += (size_t)N * HC;   // GEMM output per layer (later reused as z128)
  float* bufA  = ws;  ws += (size_t)N * HC;   // aggregated features (layers 0 & 2)
  float* bufB  = ws;  ws += (size_t)N * HC;   // aggregated features (layer 1; later y64)
  float* a_src = ws;  ws += (size_t)N * HEADS;
  float* a_dst = ws;  ws += (size_t)N * HEADS;
  float* mbuf  = ws;  ws += (size_t)N * HEADS;
  float* sbuf  = ws;  ws += (size_t)N * HEADS;

  const int nh      = N * HEADS;
  const int attnBlk = (nh * 32 + 255) / 256;
  const int nhBlk   = (nh + 255) / 256;
  const int nhcBlk  = (N * HC + 255) / 256;
  const int edgeBlk = (Etot * HEADS + 255) / 256;
  const int aggBlk  = (Etot * 32 + 255) / 256;
  const dim3 gemmGrid((N + 15) / 16, HC / 128);

  // attention + segment softmax + scatter for one layer
  auto edge_phase = [&](const float* as_, const float* ad_, const float* bias_,
                        float* out_) {
    attn_coeff<<<attnBlk, 256, 0, stream>>>(h_lin, as_, ad_, a_src, a_dst, N);
    init_attn_buffers<<<nhBlk, 256, 0, stream>>>(mbuf, sbuf, nh);
    init_out_bias<<<nhcBlk, 256, 0, stream>>>(out_, bias_, N * HC);
    edge_max<<<edgeBlk, 256, 0, stream>>>(ei, E, Etot, a_src, a_dst, mbuf);
    edge_expsum<<<edgeBlk, 256, 0, stream>>>(ei, E, Etot, a_src, a_dst, mbuf, sbuf);
    edge_aggregate<<<aggBlk, 256, 0, stream>>>(ei, E, Etot, a_src, a_dst, mbuf, sbuf,
                                               h_lin, out_);
  };

  // ---- layer 0: x[N,128] @ W0 -> h, attention, aggregate (+b0) -> bufA
  wmma_gemm_f32<128, HC><<<gemmGrid, 256, 0, stream>>>(x, W0, h_lin, N, nullptr, 0);
  edge_phase(as0, ad0, b0, bufA);

  // ---- layer 1: bufA @ W1 -> h, aggregate (+b1) -> bufB
  wmma_gemm_f32<HC, HC><<<gemmGrid, 256, 0, stream>>>(bufA, W1, h_lin, N, nullptr, 0);
  edge_phase(as1, ad1, b1, bufB);

  // ---- layer 2: bufB @ W2 -> h, aggregate (no bias; concat=False) -> bufA
  wmma_gemm_f32<HC, HC><<<gemmGrid, 256, 0, stream>>>(bufB, W2, h_lin, N, nullptr, 0);
  edge_phase(as2, ad2, nullptr, bufA);

  // mean over heads + b2  -> y64 (reuse bufB)
  float* y64 = bufB;
  head_mean_bias<<<(N * CH + 255) / 256, 256, 0, stream>>>(bufA, b2, y64, N);

  // relu(y64 @ lw0 + lb0) -> z128 (reuse h_lin)   [N,64]@[64,128]
  float* z128 = h_lin;
  wmma_gemm_f32<CH, 128><<<dim3((N + 15) / 16, 1), 256, 0, stream>>>(y64, lw0, z128, N,
                                                                     lb0, 1);

  // z128 @ lw1 + lb1 -> d_out  [N,10]
  final_linear<<<(N * 10 + 255) / 256, 256, 0, stream>>>(z128, lw1, lb1,
                                                         (float*)d_out, N);
}